// DualEdgeConv_14242111554012
// MI455X (gfx1250) — compile-verified
//
#include <hip/hip_runtime.h>

typedef __attribute__((ext_vector_type(2))) float v2f;
typedef __attribute__((ext_vector_type(8))) float v8f;

#define DEV __device__ __forceinline__

constexpr int B  = 8;
constexpr int N  = 4096;
constexpr int KN = 16;   // k neighbors
constexpr int CP = 64;   // pos channels
constexpr int CE = 32;   // energy channels

// ---------------- CDNA5 fp32 WMMA: D(16x16) += A(16x4) x B(4x16) -------------
DEV v8f wmma4(v2f a, v2f b, v8f c) {
  // 8-arg form: (neg_a, A, neg_b, B, c_mod, C, reuse_a, reuse_b)
  return __builtin_amdgcn_wmma_f32_16x16x4_f32(false, a, false, b, (short)0, c,
                                               false, false);
}

// ---------------------------------------------------------------------------
// Kernel 1: brute-force kNN via WMMA distance tiles + per-row top-16.
// grid=(N/16, B), block=32 (one wave).  idxOut: (B,N,K) int32
// ---------------------------------------------------------------------------
__global__ __launch_bounds__(32) void knn_kernel(const float* __restrict__ pos,
                                                 int* __restrict__ idxOut) {
  const int b    = blockIdx.y;
  const int i0   = blockIdx.x * 16;
  const int lane = threadIdx.x;
  const float* posB = pos + (size_t)b * N * CP;

  __shared__ float Arow[16][CP];
  __shared__ float Brow[16][CP];
  __shared__ float sqA[16];
  __shared__ float sqB[16];
  __shared__ float distT[16][16];
  __shared__ float bestD[16][KN];
  __shared__ int   bestI[16][KN];

  {  // load 16-row A tile (each lane: half a row)
    int r = lane >> 1, h = lane & 1;
    const float4* s = (const float4*)(posB + (size_t)(i0 + r) * CP + h * 32);
    float4* d = (float4*)(&Arow[r][h * 32]);
#pragma unroll
    for (int q = 0; q < 8; ++q) d[q] = s[q];
  }
  __syncthreads();
  if (lane < 16) {
    float s = 0.f;
    for (int c = 0; c < CP; ++c) { float v = Arow[lane][c]; s += v * v; }
    sqA[lane] = s;
#pragma unroll
    for (int p = 0; p < KN; ++p) { bestD[lane][p] = 3.0e38f; bestI[lane][p] = 0; }
  }
  __syncthreads();

  const int m16  = lane & 15;
  const int koff = (lane >> 4) << 1;   // 0 or 2 (A/B fragment K select)
  const int mofs = (lane >> 4) << 3;   // 0 or 8 (C/D row half)

  for (int jt = 0; jt < N / 16; ++jt) {
    const int j0 = jt * 16;
    {  // load B tile
      int r = lane >> 1, h = lane & 1;
      const float4* s = (const float4*)(posB + (size_t)(j0 + r) * CP + h * 32);
      float4* d = (float4*)(&Brow[r][h * 32]);
#pragma unroll
      for (int q = 0; q < 8; ++q) d[q] = s[q];
    }
    __syncthreads();
    if (lane < 16) {
      float s = 0.f;
      for (int c = 0; c < CP; ++c) { float v = Brow[lane][c]; s += v * v; }
      sqB[lane] = s;
    }
    __syncthreads();

    v8f acc = {};
#pragma unroll
    for (int c0 = 0; c0 < CP; c0 += 4) {
      v2f a;  a.x  = Arow[m16][c0 + koff]; a.y  = Arow[m16][c0 + koff + 1];
      v2f bb; bb.x = Brow[m16][c0 + koff]; bb.y = Brow[m16][c0 + koff + 1];
      acc = wmma4(a, bb, acc);   // acc[m][n] = dot(x_{i0+m}, x_{j0+n})
    }

#pragma unroll
    for (int r = 0; r < 8; ++r) {
      int m = r + mofs, n = m16;
      float d2 = sqA[m] + sqB[n] - 2.f * acc[r];
      float d  = sqrtf(fmaxf(d2, 0.f));
      if (i0 + m == j0 + n) d += 1e9f;   // mask diagonal
      distT[m][n] = d;
    }
    __syncthreads();

    if (lane < 16) {  // per-row sorted insertion into top-16
      int m = lane;
      for (int n = 0; n < 16; ++n) {
        float d = distT[m][n];
        if (d < bestD[m][KN - 1]) {
          int p = KN - 1;
          while (p > 0 && bestD[m][p - 1] > d) {
            bestD[m][p] = bestD[m][p - 1];
            bestI[m][p] = bestI[m][p - 1];
            --p;
          }
          bestD[m][p] = d;
          bestI[m][p] = j0 + n;
        }
      }
    }
    __syncthreads();
  }

  if (lane < 16) {
#pragma unroll
    for (int p = 0; p < KN; ++p)
      idxOut[((size_t)b * N + (i0 + lane)) * KN + p] = bestI[lane][p];
  }
}

// ---------------------------------------------------------------------------
// Shared helpers for the 16x64 edge-feature GEMMs
// ---------------------------------------------------------------------------
DEV void load_w64(const float* __restrict__ W, float* lds, int lane) {
  const float4* s = (const float4*)W;
  float4* d = (float4*)lds;
  for (int q = lane; q < (64 * 64) / 4; q += 32) d[q] = s[q];
}

DEV void build_dx(const float* __restrict__ posB, const int* __restrict__ idxP,
                  const float* xi, float (*dx)[64], int lane) {
  int k = lane >> 1, h = lane & 1;
  int j = idxP[k];
  const float* pj = posB + (size_t)j * CP + h * 32;
  const float* pi = xi + h * 32;
#pragma unroll
  for (int c = 0; c < 32; ++c) dx[k][h * 32 + c] = pj[c] - pi[c];
}

// h(16x64) = A(16x64) @ W(64x64) + bias ; accs[ot] holds cols [16*ot,16*ot+16)
DEV void gemm16x64(const float (*A)[64], const float* __restrict__ Wl,
                   const float* __restrict__ bias, v8f accs[4], int lane) {
  const int m16  = lane & 15;
  const int koff = (lane >> 4) << 1;
#pragma unroll
  for (int ot = 0; ot < 4; ++ot) accs[ot] = (v8f){};
#pragma unroll
  for (int c0 = 0; c0 < 64; c0 += 4) {
    v2f a; a.x = A[m16][c0 + koff]; a.y = A[m16][c0 + koff + 1];
#pragma unroll
    for (int ot = 0; ot < 4; ++ot) {
      v2f bb;
      bb.x = Wl[(c0 + koff) * 64 + ot * 16 + m16];
      bb.y = Wl[(c0 + koff + 1) * 64 + ot * 16 + m16];
      accs[ot] = wmma4(a, bb, accs[ot]);
    }
  }
#pragma unroll
  for (int ot = 0; ot < 4; ++ot) {
    float bc = bias[ot * 16 + m16];
#pragma unroll
    for (int r = 0; r < 8; ++r) accs[ot][r] += bc;
  }
}

// ---------------------------------------------------------------------------
// Kernel 2 (pass 1): accumulate per-channel sum / sumsq of h = dx@Wx1+bx1
// grid=(N/16, B), block=32; each block processes 16 points (16 row-tiles).
// ---------------------------------------------------------------------------
__global__ __launch_bounds__(32) void edge_stats_kernel(
    const float* __restrict__ pos, const int* __restrict__ idx,
    const float* __restrict__ Wx1, const float* __restrict__ bx1,
    float* __restrict__ stat) {
  __shared__ float Wl[64 * 64];
  __shared__ float dx[16][64];
  __shared__ float xi[64];
  __shared__ float lsum[64];
  __shared__ float lsq[64];

  const int lane = threadIdx.x;
  const int b    = blockIdx.y;
  const float* posB = pos + (size_t)b * N * CP;

  load_w64(Wx1, Wl, lane);
  for (int c = lane; c < 64; c += 32) { lsum[c] = 0.f; lsq[c] = 0.f; }
  __syncthreads();

  for (int p = 0; p < 16; ++p) {
    const int n = blockIdx.x * 16 + p;
    if (lane < 16)
      ((float4*)xi)[lane] = ((const float4*)(posB + (size_t)n * CP))[lane];
    __syncthreads();
    build_dx(posB, idx + ((size_t)b * N + n) * KN, xi, dx, lane);
    __syncthreads();

    v8f accs[4];
    gemm16x64(dx, Wl, bx1, accs, lane);

    const int m16 = lane & 15;
#pragma unroll
    for (int ot = 0; ot < 4; ++ot) {
      float s = 0.f, q = 0.f;
#pragma unroll
      for (int r = 0; r < 8; ++r) { float v = accs[ot][r]; s += v; q += v * v; }
      atomicAdd(&lsum[ot * 16 + m16], s);   // ds_add_f32
      atomicAdd(&lsq[ot * 16 + m16], q);
    }
    __syncthreads();
  }
  for (int c = lane; c < 64; c += 32) {
    atomicAdd(&stat[c], lsum[c]);
    atomicAdd(&stat[64 + c], lsq[c]);
  }
}

__global__ void zero_stats(float* stat) { stat[threadIdx.x] = 0.f; }

__global__ void finalize_stats(const float* __restrict__ gx,
                               const float* __restrict__ betax,
                               float* __restrict__ stat) {
  int c = threadIdx.x;  // 64 threads
  const float invR = 1.f / (float)((size_t)B * N * KN);
  float mean = stat[c] * invR;
  float var  = stat[64 + c] * invR - mean * mean;
  float sc   = gx[c] * rsqrtf(var + 1e-5f);
  stat[128 + c] = sc;                 // scale
  stat[192 + c] = betax[c] - mean * sc;  // shift
}

// ---------------------------------------------------------------------------
// Kernel 4 (pass 2): h1=BN+ReLU(dx@Wx1+bx1); h2=h1@Wx2+bx2; max_k; + x@Wpx+bpx
// grid=(N/16, B), block=32; per block: 16 points; phi done as one WMMA tile.
// ---------------------------------------------------------------------------
__global__ __launch_bounds__(32) void pos_out_kernel(
    const float* __restrict__ pos, const int* __restrict__ idx,
    const float* __restrict__ Wx1, const float* __restrict__ bx1,
    const float* __restrict__ Wx2, const float* __restrict__ bx2,
    const float* __restrict__ Wpx, const float* __restrict__ bpx,
    const float* __restrict__ stat, float* __restrict__ posOut) {
  __shared__ float W1[64 * 64];
  __shared__ float W2[64 * 64];          // Wpx first, then Wx2
  __shared__ float dxT[16][64];          // also the X tile for phi
  __shared__ float hT[16][64];
  __shared__ float phiT[16][64];
  __shared__ float xi[64];
  __shared__ float scaleS[64];
  __shared__ float shiftS[64];

  const int lane = threadIdx.x;
  const int b    = blockIdx.y;
  const int n0   = blockIdx.x * 16;
  const float* posB = pos + (size_t)b * N * CP;
  const int m16  = lane & 15;
  const int mofs = (lane >> 4) << 3;

  // ---- phi branch: one 16-point tile through Wpx ----
  load_w64(Wpx, W2, lane);
  {
    int r = lane >> 1, h = lane & 1;
    const float4* s = (const float4*)(posB + (size_t)(n0 + r) * CP + h * 32);
    float4* d = (float4*)(&dxT[r][h * 32]);
#pragma unroll
    for (int q = 0; q < 8; ++q) d[q] = s[q];
  }
  __syncthreads();
  {
    v8f accs[4];
    gemm16x64(dxT, W2, bpx, accs, lane);
#pragma unroll
    for (int ot = 0; ot < 4; ++ot)
#pragma unroll
      for (int r = 0; r < 8; ++r) phiT[r + mofs][ot * 16 + m16] = accs[ot][r];
  }
  __syncthreads();

  // ---- weights / BN params for the edge MLP ----
  load_w64(Wx1, W1, lane);
  load_w64(Wx2, W2, lane);
  for (int c = lane; c < 64; c += 32) {
    scaleS[c] = stat[128 + c];
    shiftS[c] = stat[192 + c];
  }
  __syncthreads();

  for (int p = 0; p < 16; ++p) {
    const int n = n0 + p;
    if (lane < 16)
      ((float4*)xi)[lane] = ((const float4*)(posB + (size_t)n * CP))[lane];
    __syncthreads();
    build_dx(posB, idx + ((size_t)b * N + n) * KN, xi, dxT, lane);
    __syncthreads();

    v8f accs[4];
    gemm16x64(dxT, W1, bx1, accs, lane);   // layer 1 + bias
#pragma unroll
    for (int ot = 0; ot < 4; ++ot) {       // BN + ReLU -> LDS
      int col = ot * 16 + m16;
      float sc = scaleS[col], sh = shiftS[col];
#pragma unroll
      for (int r = 0; r < 8; ++r)
        hT[r + mofs][col] = fmaxf(accs[ot][r] * sc + sh, 0.f);
    }
    __syncthreads();

    v8f acc2[4];
    gemm16x64(hT, W2, bx2, acc2, lane);    // layer 2 + bias
#pragma unroll
    for (int ot = 0; ot < 4; ++ot) {       // max over the 16 neighbor rows
      float mx = acc2[ot][0];
#pragma unroll
      for (int r = 1; r < 8; ++r) mx = fmaxf(mx, acc2[ot][r]);
      mx = fmaxf(mx, __shfl_xor(mx, 16, 32));  // rows 0-7 vs 8-15
      if (lane < 16) {
        int col = ot * 16 + lane;
        posOut[((size_t)b * N + n) * CP + col] = mx + phiT[p][col];
      }
    }
    __syncthreads();
  }
}

// ---------------------------------------------------------------------------
// Kernel 5: energy branch.  mean_k(de@Wte+bte) = ((mean_k e_j)-e_i)@Wte + bte
// grid=(N/16, B), block=32.
// ---------------------------------------------------------------------------
__global__ __launch_bounds__(32) void energy_kernel(
    const float* __restrict__ ene, const int* __restrict__ idx,
    const float* __restrict__ Wte, const float* __restrict__ bte,
    const float* __restrict__ Wpe, const float* __restrict__ bpe,
    float* __restrict__ eneOut) {
  __shared__ float WeL[32 * 32];
  __shared__ float WpL[32 * 32];
  __shared__ float A1[16][32];   // mean_k(e_j) - e_i
  __shared__ float A2[16][32];   // e_i

  const int lane = threadIdx.x;
  const int b    = blockIdx.y;
  const int n0   = blockIdx.x * 16;
  const float* eB = ene + (size_t)b * N * CE;

  {
    const float4* s1 = (const float4*)Wte;
    const float4* s2 = (const float4*)Wpe;
    float4* d1 = (float4*)WeL;
    float4* d2 = (float4*)WpL;
    for (int q = lane; q < (32 * 32) / 4; q += 32) { d1[q] = s1[q]; d2[q] = s2[q]; }
  }

  {  // build A1/A2: lane -> (point p = lane&15, channel half = lane>>4)
    int p = lane & 15, h = lane >> 4;
    int n = n0 + p;
    float s[16];
#pragma unroll
    for (int c = 0; c < 16; ++c) s[c] = 0.f;
    const int* ip = idx + ((size_t)b * N + n) * KN;
    for (int k = 0; k < KN; ++k) {
      const float* ej = eB + (size_t)ip[k] * CE + h * 16;
#pragma unroll
      for (int c = 0; c < 16; ++c) s[c] += ej[c];
    }
    const float* ei = eB + (size_t)n * CE + h * 16;
#pragma unroll
    for (int c = 0; c < 16; ++c) {
      float e = ei[c];
      A1[p][h * 16 + c] = s[c] * (1.f / (float)KN) - e;
      A2[p][h * 16 + c] = e;
    }
  }
  __syncthreads();

  const int m16  = lane & 15;
  const int koff = (lane >> 4) << 1;
  const int mofs = (lane >> 4) << 3;
#pragma unroll
  for (int ot = 0; ot < 2; ++ot) {
    v8f acc = {};
#pragma unroll
    for (int c0 = 0; c0 < 32; c0 += 4) {
      v2f a;  a.x  = A1[m16][c0 + koff];           a.y  = A1[m16][c0 + koff + 1];
      v2f bb; bb.x = WeL[(c0 + koff) * 32 + ot * 16 + m16];
              bb.y = WeL[(c0 + koff + 1) * 32 + ot * 16 + m16];
      acc = wmma4(a, bb, acc);
    }
#pragma unroll
    for (int c0 = 0; c0 < 32; c0 += 4) {
      v2f a;  a.x  = A2[m16][c0 + koff];           a.y  = A2[m16][c0 + koff + 1];
      v2f bb; bb.x = WpL[(c0 + koff) * 32 + ot * 16 + m16];
              bb.y = WpL[(c0 + koff + 1) * 32 + ot * 16 + m16];
      acc = wmma4(a, bb, acc);
    }
    const int col = ot * 16 + m16;
    const float badd = bte[col] + bpe[col];
#pragma unroll
    for (int r = 0; r < 8; ++r) {
      int m = r + mofs;
      eneOut[((size_t)b * N + (n0 + m)) * CE + col] = acc[r] + badd;
    }
  }
}

// ---------------------------------------------------------------------------
extern "C" void kernel_launch(void* const* d_in, const int* in_sizes, int n_in,
                              void* d_out, int out_size, void* d_ws,
                              size_t ws_size, hipStream_t stream) {
  (void)in_sizes; (void)n_in; (void)out_size; (void)ws_size;
  const float* pos   = (const float*)d_in[0];
  const float* ene   = (const float*)d_in[1];
  const float* Wx1   = (const float*)d_in[2];
  const float* bx1   = (const float*)d_in[3];
  const float* gx    = (const float*)d_in[4];
  const float* betax = (const float*)d_in[5];
  const float* Wx2   = (const float*)d_in[6];
  const float* bx2   = (const float*)d_in[7];
  const float* Wpx   = (const float*)d_in[8];
  const float* bpx   = (const float*)d_in[9];
  const float* Wte   = (const float*)d_in[10];
  const float* bte   = (const float*)d_in[11];
  const float* Wpe   = (const float*)d_in[12];
  const float* bpe   = (const float*)d_in[13];
  // d_in[14] = k (16), compile-time constant here.

  int*   idxWs  = (int*)d_ws;
  float* stat   = (float*)((char*)d_ws + (size_t)B * N * KN * sizeof(int));
  float* posOut = (float*)d_out;
  float* eneOut = posOut + (size_t)B * N * CP;

  knn_kernel<<<dim3(N / 16, B), 32, 0, stream>>>(pos, idxWs);
  zero_stats<<<1, 128, 0, stream>>>(stat);
  edge_stats_kernel<<<dim3(N / 16, B), 32, 0, stream>>>(pos, idxWs, Wx1, bx1,
                                                        stat);
  finalize_stats<<<1, 64, 0, stream>>>(gx, betax, stat);
  pos_out_kernel<<<dim3(N / 16, B), 32, 0, stream>>>(
      pos, idxWs, Wx1, bx1, Wx2, bx2, Wpx, bpx, stat, posOut);
  energy_kernel<<<dim3(N / 16, B), 32, 0, stream>>>(ene, idxWs, Wte, bte, Wpe,
                                                    bpe, eneOut);
}